// PhysicsInformedNN_40278203301974
// MI455X (gfx1250) — compile-verified
//
#include <hip/hip_runtime.h>

typedef float v2f __attribute__((ext_vector_type(2)));
typedef float v8f __attribute__((ext_vector_type(8)));

#define HID   20
#define NCH   13
#define NLAY  8
#define WPB   2                 // waves per block (wave32)
#define SPW   16                // samples per wave (one WMMA N-tile)
#define SPB   (WPB * SPW)
#define HCOLS (NCH * SPW)       // 208 columns of jet state

// ---------------------------------------------------------------------------
// tanh jet chain rule: given pre-activation jet a[13], produce activation jet.
// channels: 0=val 1=dx 2=dy 3=dt 4=xx 5=xy 6=xt 7=yy 8=yt 9=xxx 10=xxy 11=xyy 12=yyy
// ---------------------------------------------------------------------------
__device__ __forceinline__ void tanh_chain(const float a[NCH], float h[NCH]) {
    float t  = tanhf(a[0]);
    float s1 = 1.0f - t * t;
    float s2 = -2.0f * t * s1;
    float s3 = 2.0f * s1 * (2.0f * t * t - s1);
    h[0] = t;
    h[1] = s1 * a[1];
    h[2] = s1 * a[2];
    h[3] = s1 * a[3];
    h[4] = s1 * a[4] + s2 * a[1] * a[1];
    h[5] = s1 * a[5] + s2 * a[1] * a[2];
    h[6] = s1 * a[6] + s2 * a[1] * a[3];
    h[7] = s1 * a[7] + s2 * a[2] * a[2];
    h[8] = s1 * a[8] + s2 * a[2] * a[3];
    h[9]  = s1 * a[9]  + 3.0f * s2 * a[4] * a[1] + s3 * a[1] * a[1] * a[1];
    h[10] = s1 * a[10] + s2 * (a[4] * a[2] + 2.0f * a[5] * a[1]) + s3 * a[1] * a[1] * a[2];
    h[11] = s1 * a[11] + s2 * (a[7] * a[1] + 2.0f * a[5] * a[2]) + s3 * a[1] * a[2] * a[2];
    h[12] = s1 * a[12] + 3.0f * s2 * a[7] * a[2] + s3 * a[2] * a[2] * a[2];
}

// ---------------------------------------------------------------------------
// global min / max of xyt (lb / ub), two-pass deterministic tree reduction
// ws layout (floats): [0..63]=block mins, [64..127]=block maxs, [128]=lb, [129]=ub
// ---------------------------------------------------------------------------
__global__ void minmax_part(const float* __restrict__ x, int n, float* __restrict__ ws) {
    __shared__ float smin[256], smax[256];
    float mn = 3.4e38f, mx = -3.4e38f;
    for (int i = blockIdx.x * blockDim.x + threadIdx.x; i < n; i += gridDim.x * blockDim.x) {
        float v = x[i];
        mn = fminf(mn, v);
        mx = fmaxf(mx, v);
    }
    smin[threadIdx.x] = mn; smax[threadIdx.x] = mx;
    __syncthreads();
    for (int s = 128; s > 0; s >>= 1) {
        if ((int)threadIdx.x < s) {
            smin[threadIdx.x] = fminf(smin[threadIdx.x], smin[threadIdx.x + s]);
            smax[threadIdx.x] = fmaxf(smax[threadIdx.x], smax[threadIdx.x + s]);
        }
        __syncthreads();
    }
    if (threadIdx.x == 0) { ws[blockIdx.x] = smin[0]; ws[64 + blockIdx.x] = smax[0]; }
}

__global__ void minmax_final(float* __restrict__ ws) {
    __shared__ float smin[64], smax[64];
    int t = threadIdx.x;
    smin[t] = ws[t]; smax[t] = ws[64 + t];
    __syncthreads();
    for (int s = 32; s > 0; s >>= 1) {
        if (t < s) {
            smin[t] = fminf(smin[t], smin[t + s]);
            smax[t] = fmaxf(smax[t], smax[t + s]);
        }
        __syncthreads();
    }
    if (t == 0) { ws[128] = smin[0]; ws[129] = smax[0]; }
}

// ---------------------------------------------------------------------------
// main PINN kernel: 13-channel jet propagation, fp32 WMMA 16x16x4 GEMMs
// ---------------------------------------------------------------------------
__global__ __launch_bounds__(WPB * 32)
void pinn_wmma_kernel(const float* __restrict__ xyt,
                      const float* __restrict__ W0, const float* __restrict__ b0,
                      const float* __restrict__ Wh, const float* __restrict__ bh,
                      const float* __restrict__ Wo, const float* __restrict__ bo,
                      const float* __restrict__ lam1, const float* __restrict__ lam2,
                      const float* __restrict__ lbub,
                      float* __restrict__ out, int nsamp)
{
    __shared__ float Hbuf[WPB][HID][HCOLS];   // per-wave jet state, 33.3 KB
    __shared__ float Wbuf[32][HID];           // layer weights, rows 20..31 zero-pad
    __shared__ float Bbuf[HID];
    __shared__ float W0b[HID][3];
    __shared__ float b0b[HID];
    __shared__ float Zb[WPB][3][SPW];

    const int tid   = threadIdx.x;
    const int wv    = tid >> 5;
    const int lane  = tid & 31;
    const int hi    = lane >> 4;       // half-wave select (WMMA fragment layout)
    const int lm    = lane & 15;
    const int sbase = blockIdx.x * SPB + wv * SPW;

    // one-time zero pad of weight rows 20..31, stage first-layer params
    for (int i = tid; i < 12 * HID; i += blockDim.x) Wbuf[20 + i / HID][i % HID] = 0.0f;
    for (int i = tid; i < HID * 3; i += blockDim.x)  W0b[i / 3][i % 3] = W0[i];
    for (int i = tid; i < HID; i += blockDim.x)      b0b[i] = b0[i];

    const float lb = lbub[0], ub = lbub[1];
    const float alpha = 2.0f / (ub - lb);
    const float beta  = -(lb + ub) / (ub - lb);

    if (hi == 0) {
        const int s = sbase + lm;
        Zb[wv][0][lm] = alpha * xyt[3 * s + 0] + beta;
        Zb[wv][1][lm] = alpha * xyt[3 * s + 1] + beta;
        Zb[wv][2][lm] = alpha * xyt[3 * s + 2] + beta;
    }
    __syncthreads();

    // ---- layer 1 (3 -> HID), VALU: fills the jet state -----------------
    // pre-activation jet: a0 = W0 z + b0, a_x = alpha*W0[:,0], etc.; 2nd/3rd = 0
    for (int it = 0; it < (HID * SPW) / 32; ++it) {
        const int p = it * 32 + lane;
        const int n = p >> 4;          // neuron
        const int c = p & 15;          // sample column
        const float w0 = W0b[n][0], w1 = W0b[n][1], w2 = W0b[n][2];
        float a[NCH] = {0};
        a[0] = b0b[n] + w0 * Zb[wv][0][c] + w1 * Zb[wv][1][c] + w2 * Zb[wv][2][c];
        a[1] = alpha * w0;
        a[2] = alpha * w1;
        a[3] = alpha * w2;
        float h[NCH];
        tanh_chain(a, h);
        #pragma unroll
        for (int ch = 0; ch < NCH; ++ch) Hbuf[wv][n][ch * SPW + c] = h[ch];
    }

    // ---- hidden layers: WMMA f32 16x16x4 GEMM + tanh jet ----------------
    for (int L = 0; L < NLAY; ++L) {
        __syncthreads();
        for (int i = tid; i < HID * HID; i += blockDim.x)
            Wbuf[i / HID][i % HID] = Wh[L * HID * HID + i];
        for (int i = tid; i < HID; i += blockDim.x) Bbuf[i] = bh[L * HID + i];
        __syncthreads();

        v8f   acc0[NCH];        // rows 0..15
        float acc1[NCH][4];     // rows 16..19 (useful part of second M-tile)
        #pragma unroll
        for (int ch = 0; ch < NCH; ++ch) {
            v8f c0 = {0, 0, 0, 0, 0, 0, 0, 0};
            v8f c1 = {0, 0, 0, 0, 0, 0, 0, 0};
            #pragma unroll
            for (int k = 0; k < HID; k += 4) {
                v2f bfr, af0, af1;
                bfr.x = Hbuf[wv][k + 2 * hi + 0][ch * SPW + lm];
                bfr.y = Hbuf[wv][k + 2 * hi + 1][ch * SPW + lm];
                af0.x = Wbuf[lm][k + 2 * hi + 0];
                af0.y = Wbuf[lm][k + 2 * hi + 1];
                af1.x = Wbuf[16 + lm][k + 2 * hi + 0];
                af1.y = Wbuf[16 + lm][k + 2 * hi + 1];
                c0 = __builtin_amdgcn_wmma_f32_16x16x4_f32(false, af0, false, bfr,
                                                           (short)0, c0, false, false);
                c1 = __builtin_amdgcn_wmma_f32_16x16x4_f32(false, af1, false, bfr,
                                                           (short)0, c1, false, false);
            }
            acc0[ch] = c0;
            acc1[ch][0] = c1[0]; acc1[ch][1] = c1[1];
            acc1[ch][2] = c1[2]; acc1[ch][3] = c1[3];
        }

        // nonlinearity element-wise on the C fragments; write back jet state
        #pragma unroll
        for (int r = 0; r < 8; ++r) {
            const int row = hi ? (8 + r) : r;
            float a[NCH], h[NCH];
            #pragma unroll
            for (int ch = 0; ch < NCH; ++ch) a[ch] = acc0[ch][r];
            a[0] += Bbuf[row];
            tanh_chain(a, h);
            #pragma unroll
            for (int ch = 0; ch < NCH; ++ch) Hbuf[wv][row][ch * SPW + lm] = h[ch];
        }
        #pragma unroll
        for (int r = 0; r < 4; ++r) {
            const int row = 16 + r;
            float a[NCH], h[NCH];
            #pragma unroll
            for (int ch = 0; ch < NCH; ++ch) a[ch] = acc1[ch][r];
            a[0] += Bbuf[row];
            tanh_chain(a, h);
            if (hi == 0) {
                #pragma unroll
                for (int ch = 0; ch < NCH; ++ch) Hbuf[wv][row][ch * SPW + lm] = h[ch];
            }
        }
    }

    // ---- output layer (HID -> 2), no tanh: WMMA with zero-padded rows ---
    __syncthreads();
    for (int i = tid; i < 2 * HID; i += blockDim.x)  Wbuf[i / HID][i % HID] = Wo[i];
    for (int i = tid; i < 18 * HID; i += blockDim.x) Wbuf[2 + i / HID][i % HID] = 0.0f;
    __syncthreads();

    v8f oacc[NCH];
    #pragma unroll
    for (int ch = 0; ch < NCH; ++ch) {
        v8f c0 = {0, 0, 0, 0, 0, 0, 0, 0};
        #pragma unroll
        for (int k = 0; k < HID; k += 4) {
            v2f bfr, afr;
            bfr.x = Hbuf[wv][k + 2 * hi + 0][ch * SPW + lm];
            bfr.y = Hbuf[wv][k + 2 * hi + 1][ch * SPW + lm];
            afr.x = Wbuf[lm][k + 2 * hi + 0];
            afr.y = Wbuf[lm][k + 2 * hi + 1];
            c0 = __builtin_amdgcn_wmma_f32_16x16x4_f32(false, afr, false, bfr,
                                                       (short)0, c0, false, false);
        }
        oacc[ch] = c0;
    }

    // row 0 of D = psi jet, row 1 = p jet; valid in lanes 0..15 (VGPR 0/1)
    const float l1  = lam1[0];
    const float l2  = lam2[0];
    const float bo1 = bo[1];
    if (hi == 0) {
        const int s = sbase + lm;
        const float u    =  oacc[2][0];          // psi_y
        const float vv   = -oacc[1][0];          // -psi_x
        const float p    =  oacc[0][1] + bo1;    // p value (+ bias)
        const float p_x  =  oacc[1][1];
        const float p_y  =  oacc[2][1];
        const float u_x  =  oacc[5][0];          // psi_xy
        const float u_y  =  oacc[7][0];          // psi_yy
        const float u_t  =  oacc[8][0];          // psi_yt
        const float v_x  = -oacc[4][0];          // -psi_xx
        const float v_y  = -oacc[5][0];          // -psi_xy
        const float v_t  = -oacc[6][0];          // -psi_xt
        const float u_xx =  oacc[10][0];         // psi_xxy
        const float u_yy =  oacc[12][0];         // psi_yyy
        const float v_xx = -oacc[9][0];          // -psi_xxx
        const float v_yy = -oacc[11][0];         // -psi_xyy
        const float f_u = u_t + l1 * (u * u_x + vv * u_y) + p_x - l2 * (u_xx + u_yy);
        const float f_v = v_t + l1 * (u * v_x + vv * v_y) + p_y - l2 * (v_xx + v_yy);
        out[0 * nsamp + s] = u;
        out[1 * nsamp + s] = vv;
        out[2 * nsamp + s] = p;
        out[3 * nsamp + s] = f_u;
        out[4 * nsamp + s] = f_v;
    }
}

// ---------------------------------------------------------------------------
extern "C" void kernel_launch(void* const* d_in, const int* in_sizes, int n_in,
                              void* d_out, int out_size, void* d_ws, size_t ws_size,
                              hipStream_t stream) {
    const float* xyt  = (const float*)d_in[0];
    const float* W0   = (const float*)d_in[1];
    const float* b0   = (const float*)d_in[2];
    const float* Wh   = (const float*)d_in[3];
    const float* bh   = (const float*)d_in[4];
    const float* Wo   = (const float*)d_in[5];
    const float* bo   = (const float*)d_in[6];
    const float* lam1 = (const float*)d_in[7];
    const float* lam2 = (const float*)d_in[8];
    float* ws  = (float*)d_ws;
    float* out = (float*)d_out;

    const int ntot  = in_sizes[0];       // 3 * N
    const int nsamp = ntot / 3;          // N = 65536

    minmax_part<<<64, 256, 0, stream>>>(xyt, ntot, ws);
    minmax_final<<<1, 64, 0, stream>>>(ws);

    const int nblocks = nsamp / SPB;     // 2048 blocks of 64 threads (2 waves)
    pinn_wmma_kernel<<<nblocks, WPB * 32, 0, stream>>>(
        xyt, W0, b0, Wh, bh, Wo, bo, lam1, lam2, ws + 128, out, nsamp);
}